// GATModel_1151051235495
// MI455X (gfx1250) — compile-verified
//
#include <hip/hip_runtime.h>
#include <math.h>

// ---------------------------------------------------------------------------
// GAT forward for MI455X (gfx1250, wave32).
// GEMMs use V_WMMA_F32_16X16X4_F32 (full fp32 precision, matches reference).
// Edge softmax/scatter via f32 atomics -- h/out buffers (25.6MB each) are
// L2-resident (192MB L2), so random gather/scatter stays on-chip.
// ---------------------------------------------------------------------------

typedef __attribute__((ext_vector_type(2))) float v2f;
typedef __attribute__((ext_vector_type(8))) float v8f;

#define FIN 128   // input features
#define HC  128   // H*C
#define GG  512   // num_graphs (reference constant)
#define LAT 32    // fc output
#define NEG_SLOPE 0.2f
#define BN_EPS 1e-5f

#define WMMA_F32(a, b, c) \
  __builtin_amdgcn_wmma_f32_16x16x4_f32(false, (a), false, (b), (short)0, (c), false, false)

// --- monotonic float<->uint encoding for atomicMax-based segment max --------
__device__ __forceinline__ unsigned encF(float f) {
  int i = __float_as_int(f);
  return (i < 0) ? ~(unsigned)i : ((unsigned)i | 0x80000000u);
}
__device__ __forceinline__ float decF(unsigned u) {
  int i = (u & 0x80000000u) ? (int)(u & 0x7FFFFFFFu) : ~(int)u;
  return __int_as_float(i);
}
#define ENC_NEG_INF 0x007FFFFFu  // encF(-inf)

// --- workspace init (harness poisons ws once; we must re-init every call) ---
__global__ void k_init(unsigned* __restrict__ mEnc, float* __restrict__ sSum,
                       float* __restrict__ outMsg, float* __restrict__ pooled,
                       int n2, int nHC) {
  int idx = blockIdx.x * blockDim.x + threadIdx.x;
  if (idx < n2) {
    mEnc[idx] = ENC_NEG_INF;
    sSum[idx] = 0.f;
  } else if (idx < n2 + nHC) {
    outMsg[idx - n2] = 0.f;
  } else if (idx < n2 + nHC + GG * HC) {
    pooled[idx - n2 - nHC] = 0.f;
  }
}

// --- h = x @ lin_w^T  via fp32 WMMA ----------------------------------------
// Block = 128 threads (4 waves). Block computes rows [m0, m0+32), all 128 cols.
// Wave w owns columns [w*32, w*32+32): a 2x2 grid of 16x16 tiles.
// 4 WMMAs per (2 A-frag + 2 B-frag) loads -> 1.0 WMMA per b64 load.
// A frag (16x4 f32): lane l, vgpr j -> A[l&15][k + j + 2*(l>>4)]
// B frag (4x16 f32): lane l, vgpr j -> B[k + j + 2*(l>>4)][l&15], B = W^T
// D (16x16 f32):     lane l, vgpr j -> D[j + 8*(l>>4)][l&15]
// A row indices are CLAMPED (not branched) -- row m only feeds D row m, and
// D stores are guarded, so clamped loads never pollute live output.
__global__ void k_linear(const float* __restrict__ X, const float* __restrict__ W,
                         float* __restrict__ h, int N) {
  const int lane = threadIdx.x & 31;
  const int wave = threadIdx.x >> 5;
  const int lo = lane & 15, hi = lane >> 4;
  const int m0 = blockIdx.x * 32;
  const int n0 = wave * 32;

  v8f c00 = {}, c01 = {}, c10 = {}, c11 = {};
  const int mA0 = min(m0 + lo, N - 1);
  const int mA1 = min(m0 + 16 + lo, N - 1);
  const float* xr0 = X + (size_t)mA0 * FIN;
  const float* xr1 = X + (size_t)mA1 * FIN;
  const float* w0  = W + (size_t)(n0 + lo) * FIN;
  const float* w1  = W + (size_t)(n0 + 16 + lo) * FIN;

#pragma unroll
  for (int k = 0; k < FIN; k += 4) {
    const int kk = k + 2 * hi;
    v2f a0 = *(const v2f*)(xr0 + kk);   // 8B aligned: kk even
    v2f a1 = *(const v2f*)(xr1 + kk);
    v2f b0 = *(const v2f*)(w0 + kk);
    v2f b1 = *(const v2f*)(w1 + kk);
    c00 = WMMA_F32(a0, b0, c00);
    c01 = WMMA_F32(a0, b1, c01);
    c10 = WMMA_F32(a1, b0, c10);
    c11 = WMMA_F32(a1, b1, c11);
  }
#pragma unroll
  for (int j = 0; j < 8; ++j) {
    const int ma = m0 + j + 8 * hi;        // M-tile 0
    if (ma < N) {
      h[(size_t)ma * HC + n0 + lo]      = c00[j];
      h[(size_t)ma * HC + n0 + 16 + lo] = c01[j];
    }
    const int mb = m0 + 16 + j + 8 * hi;   // M-tile 1
    if (mb < N) {
      h[(size_t)mb * HC + n0 + lo]      = c10[j];
      h[(size_t)mb * HC + n0 + 16 + lo] = c11[j];
    }
  }
}

// --- a_src[n,h] = sum_c h[n,h,c]*att_src[h,c] (same for dst) ----------------
__global__ void k_attn(const float* __restrict__ h, const float* __restrict__ attS,
                       const float* __restrict__ attD, float* __restrict__ aS,
                       float* __restrict__ aD) {
  __shared__ float ss[128], sd[128];
  const int n = blockIdx.x;
  const int t = threadIdx.x;           // t = head*64 + c
  const float v = h[(size_t)n * HC + t];
  ss[t] = v * attS[t];
  sd[t] = v * attD[t];
  __syncthreads();
  for (int off = 32; off > 0; off >>= 1) {
    if ((t & 63) < off) { ss[t] += ss[t + off]; sd[t] += sd[t + off]; }
    __syncthreads();
  }
  if ((t & 63) == 0) {
    aS[n * 2 + (t >> 6)] = ss[t];
    aD[n * 2 + (t >> 6)] = sd[t];
  }
}

// --- per-edge logits + segment max (self loops: e in [E, E+N) -> node e-E) --
__global__ void k_logits(const int* __restrict__ ei, const float* __restrict__ aS,
                         const float* __restrict__ aD, float* __restrict__ eBuf,
                         unsigned* __restrict__ mEnc, int E, int Ep) {
  const int e = blockIdx.x * blockDim.x + threadIdx.x;
  if (e >= Ep) return;
  const int src = (e < E) ? ei[e]     : (e - E);
  const int dst = (e < E) ? ei[E + e] : (e - E);
  v2f as = *(const v2f*)(aS + (size_t)src * 2);
  v2f ad = *(const v2f*)(aD + (size_t)dst * 2);
  float e0 = as.x + ad.x, e1 = as.y + ad.y;
  e0 = (e0 > 0.f) ? e0 : NEG_SLOPE * e0;
  e1 = (e1 > 0.f) ? e1 : NEG_SLOPE * e1;
  v2f ev; ev.x = e0; ev.y = e1;
  ((v2f*)eBuf)[e] = ev;
  atomicMax(mEnc + (size_t)dst * 2,     encF(e0));
  atomicMax(mEnc + (size_t)dst * 2 + 1, encF(e1));
}

// --- p = exp(e - max[dst]) ; segment sum ------------------------------------
__global__ void k_expsum(const int* __restrict__ ei, const unsigned* __restrict__ mEnc,
                         float* __restrict__ eBuf, float* __restrict__ sSum,
                         int E, int Ep) {
  const int e = blockIdx.x * blockDim.x + threadIdx.x;
  if (e >= Ep) return;
  const int dst = (e < E) ? ei[E + e] : (e - E);
  v2f ev = ((const v2f*)eBuf)[e];
  const float p0 = expf(ev.x - decF(mEnc[(size_t)dst * 2]));
  const float p1 = expf(ev.y - decF(mEnc[(size_t)dst * 2 + 1]));
  v2f pv; pv.x = p0; pv.y = p1;
  ((v2f*)eBuf)[e] = pv;
  atomicAdd(sSum + (size_t)dst * 2,     p0);
  atomicAdd(sSum + (size_t)dst * 2 + 1, p1);
}

// --- message scatter: out[dst] += h[src] * alpha. 64 threads/edge -----------
__global__ void k_scatter(const int* __restrict__ ei, const float* __restrict__ h,
                          const float* __restrict__ eBuf, const float* __restrict__ sSum,
                          float* __restrict__ outMsg, int E, int Ep) {
  const int e = blockIdx.x * 4 + (threadIdx.x >> 6);
  const int t = threadIdx.x & 63;                 // 2 channels per thread
  if (e >= Ep) return;
  const int src = (e < E) ? ei[e]     : (e - E);
  const int dst = (e < E) ? ei[E + e] : (e - E);
  const int head = t >> 5;                        // channel 2t -> head (2t)>>6
  const float alpha = eBuf[(size_t)e * 2 + head] / sSum[(size_t)dst * 2 + head];
  const int c0 = t * 2;
  v2f v = *(const v2f*)(h + (size_t)src * HC + c0);   // coalesced 512B/edge
  atomicAdd(outMsg + (size_t)dst * HC + c0,     v.x * alpha);
  atomicAdd(outMsg + (size_t)dst * HC + c0 + 1, v.y * alpha);
}

// --- pooled[batch[n]] += out[n] + bias --------------------------------------
__global__ void k_pool(const float* __restrict__ outMsg, const float* __restrict__ bias,
                       const int* __restrict__ batch, float* __restrict__ pooled, int N) {
  const int idx = blockIdx.x * blockDim.x + threadIdx.x;
  if (idx >= N * 64) return;
  const int n = idx >> 6, t = idx & 63, c0 = t * 2;
  const int g = batch[n];
  v2f v = *(const v2f*)(outMsg + (size_t)n * HC + c0);
  atomicAdd(pooled + (size_t)g * HC + c0,     v.x + bias[c0]);
  atomicAdd(pooled + (size_t)g * HC + c0 + 1, v.y + bias[c0 + 1]);
}

// --- batchnorm over G rows, per column (population var, eps=1e-5) -----------
__global__ void k_bn(const float* __restrict__ pooled, const float* __restrict__ gamma,
                     const float* __restrict__ beta, float* __restrict__ bnOut) {
  __shared__ float rs[256], rq[256];
  const int col = blockIdx.x, t = threadIdx.x;
  float s = 0.f, q = 0.f;
  for (int g = t; g < GG; g += 256) {
    const float v = pooled[(size_t)g * HC + col];
    s += v; q += v * v;
  }
  rs[t] = s; rq[t] = q;
  __syncthreads();
  for (int off = 128; off > 0; off >>= 1) {
    if (t < off) { rs[t] += rs[t + off]; rq[t] += rq[t + off]; }
    __syncthreads();
  }
  const float mu  = rs[0] / (float)GG;
  const float var = rq[0] / (float)GG - mu * mu;
  const float inv = rsqrtf(var + BN_EPS) * gamma[col];
  const float bt  = beta[col];
  for (int g = t; g < GG; g += 256) {
    bnOut[(size_t)g * HC + col] = (pooled[(size_t)g * HC + col] - mu) * inv + bt;
  }
}

// --- out[512,32] = bn @ fc_w^T + fc_b  via fp32 WMMA ------------------------
__global__ void k_fc(const float* __restrict__ A, const float* __restrict__ W,
                     const float* __restrict__ b, float* __restrict__ out) {
  const int lane = threadIdx.x & 31;
  const int wave = threadIdx.x >> 5;
  const int lo = lane & 15, hi = lane >> 4;
  const int wg = blockIdx.x * 4 + wave;   // 64 wave-tiles: 32 M-tiles x 2 N-tiles
  const int m0 = (wg >> 1) * 16;
  const int n0 = (wg & 1) * 16;

  v8f c = {};
  const float* arow = A + (size_t)(m0 + lo) * HC;
  const float* wrow = W + (size_t)(n0 + lo) * HC;
#pragma unroll
  for (int k = 0; k < HC; k += 4) {
    const int kk = k + 2 * hi;
    v2f a  = *(const v2f*)(arow + kk);
    v2f bb = *(const v2f*)(wrow + kk);
    c = WMMA_F32(a, bb, c);
  }
  const float bn_ = b[n0 + lo];
#pragma unroll
  for (int j = 0; j < 8; ++j) {
    const int m = m0 + j + 8 * hi;
    out[(size_t)m * LAT + n0 + lo] = c[j] + bn_;
  }
}

// ---------------------------------------------------------------------------
extern "C" void kernel_launch(void* const* d_in, const int* in_sizes, int n_in,
                              void* d_out, int out_size, void* d_ws, size_t ws_size,
                              hipStream_t stream) {
  const float* x     = (const float*)d_in[0];
  const int*   ei    = (const int*)d_in[1];
  const int*   batch = (const int*)d_in[2];
  // d_in[3] = num_graphs (device scalar; G=512 is a reference constant)
  const float* lin_w = (const float*)d_in[4];
  const float* attS  = (const float*)d_in[5];
  const float* attD  = (const float*)d_in[6];
  const float* bias  = (const float*)d_in[7];
  const float* gamma = (const float*)d_in[8];
  const float* beta  = (const float*)d_in[9];
  const float* fc_w  = (const float*)d_in[10];
  const float* fc_b  = (const float*)d_in[11];
  float* out = (float*)d_out;

  const int N  = in_sizes[0] / FIN;   // 50000
  const int E  = in_sizes[1] / 2;     // 1600000
  const int Ep = E + N;               // + self loops

  // workspace layout (floats); total ~ N*264 + Ep*2 + G*256 ~ 67 MB
  float*    h      = (float*)d_ws;                 // N*HC
  float*    aS     = h + (size_t)N * HC;           // N*2
  float*    aD     = aS + (size_t)N * 2;           // N*2
  unsigned* mE     = (unsigned*)(aD + (size_t)N * 2); // N*2
  float*    sSum   = (float*)(mE + (size_t)N * 2); // N*2
  float*    eBuf   = sSum + (size_t)N * 2;         // Ep*2
  float*    oMsg   = eBuf + (size_t)Ep * 2;        // N*HC
  float*    pooled = oMsg + (size_t)N * HC;        // GG*HC
  float*    bnOut  = pooled + (size_t)GG * HC;     // GG*HC

  {
    const int n2 = N * 2, nHC = N * HC;
    const int total = n2 + nHC + GG * HC;
    k_init<<<(total + 255) / 256, 256, 0, stream>>>(mE, sSum, oMsg, pooled, n2, nHC);
  }
  k_linear<<<(N + 31) / 32, 128, 0, stream>>>(x, lin_w, h, N);
  k_attn<<<N, 128, 0, stream>>>(h, attS, attD, aS, aD);
  k_logits<<<(Ep + 255) / 256, 256, 0, stream>>>(ei, aS, aD, eBuf, mE, E, Ep);
  k_expsum<<<(Ep + 255) / 256, 256, 0, stream>>>(ei, mE, eBuf, sSum, E, Ep);
  k_scatter<<<(Ep + 3) / 4, 256, 0, stream>>>(ei, h, eBuf, sSum, oMsg, E, Ep);
  {
    const int total = N * 64;
    k_pool<<<(total + 255) / 256, 256, 0, stream>>>(oMsg, bias, batch, pooled, N);
  }
  k_bn<<<HC, 256, 0, stream>>>(pooled, gamma, beta, bnOut);
  k_fc<<<16, 128, 0, stream>>>(bnOut, fc_w, fc_b, out);
}